// Attention_7816840478804
// MI455X (gfx1250) — compile-verified
//
#include <hip/hip_runtime.h>
#include <hip/hip_bf16.h>

#define N_SENT 131072
#define DIM    768
#define NBAG   16384
#define NCLS   53
#define NPAD   64

typedef float v2f __attribute__((ext_vector_type(2)));
typedef float v8f __attribute__((ext_vector_type(8)));

// ---------------------------------------------------------------------------
// Kernel 1: att[i] = dot(x[i], W[label[i]])  — one wave32 per sentence,
// float4 coalesced loads, shuffle reduction.
// ---------------------------------------------------------------------------
__global__ void att_kernel(const float* __restrict__ x, const int* __restrict__ label,
                           const float* __restrict__ W, float* __restrict__ att) {
    const int wavesPerBlock = blockDim.x >> 5;
    const int i    = blockIdx.x * wavesPerBlock + (threadIdx.x >> 5);
    const int lane = threadIdx.x & 31;
    if (i >= N_SENT) return;
    const float4* xr = (const float4*)(x + (size_t)i * DIM);
    const float4* wr = (const float4*)(W + (size_t)label[i] * DIM);
    float s = 0.f;
#pragma unroll
    for (int t = 0; t < DIM / 128; ++t) {           // 6 chunks of 128 floats
        float4 a = xr[lane + 32 * t];
        float4 b = wr[lane + 32 * t];
        s += a.x * b.x + a.y * b.y + a.z * b.z + a.w * b.w;
    }
#pragma unroll
    for (int off = 16; off > 0; off >>= 1) s += __shfl_xor(s, off, 32);
    if (lane == 0) att[i] = s;
}

// ---------------------------------------------------------------------------
// Kernel 2a/2b: segment boundary discovery (segment_ids are sorted).
// Deterministic: neighbor comparisons only, no atomics.
// ---------------------------------------------------------------------------
__global__ void seg_init_kernel(int* __restrict__ seg_start, int* __restrict__ seg_end) {
    const int b = blockIdx.x * blockDim.x + threadIdx.x;
    if (b < NBAG) { seg_start[b] = 0; seg_end[b] = 0; }
}

__global__ void seg_bounds_kernel(const int* __restrict__ seg,
                                  int* __restrict__ seg_start, int* __restrict__ seg_end) {
    const int i = blockIdx.x * blockDim.x + threadIdx.x;
    if (i >= N_SENT) return;
    const int s = seg[i];
    if (i == 0          || seg[i - 1] != s) seg_start[s] = i;
    if (i == N_SENT - 1 || seg[i + 1] != s) seg_end[s]   = i + 1;
}

// ---------------------------------------------------------------------------
// Kernel 2c: pad W (53x768 -> 64x768, zero rows beyond 53) and bias so the
// GEMM inner loop needs no EXEC predication around its B-matrix loads.
// ---------------------------------------------------------------------------
__global__ void wpad_kernel(const float* __restrict__ W, const float* __restrict__ bias,
                            float* __restrict__ wpad, float* __restrict__ bpad) {
    const int r   = blockIdx.x;                     // 0..63
    const int tid = threadIdx.x;                    // 256
    const bool valid = (r < NCLS);
    const float* src = W + (size_t)(valid ? r : 0) * DIM;
    float* dst = wpad + (size_t)r * DIM;
    for (int c = tid; c < DIM; c += 256) dst[c] = valid ? src[c] : 0.f;
    if (tid == 0) bpad[r] = valid ? bias[r] : 0.f;
}

// ---------------------------------------------------------------------------
// Kernel 3: per-bag stable softmax over att + weighted pooling of x rows.
// One 256-thread block per bag; each thread owns 3 of 768 columns.
// x is read exactly once (coalesced 256-float sweeps).
// ---------------------------------------------------------------------------
#define POOL_CHUNK 64
__global__ void pool_kernel(const float* __restrict__ x, const float* __restrict__ att,
                            const int* __restrict__ seg_start, const int* __restrict__ seg_end,
                            float* __restrict__ repre) {
    const int b   = blockIdx.x;
    const int tid = threadIdx.x;                    // 256 threads
    const int start = seg_start[b];
    const int end   = seg_end[b];

    __shared__ float s_md[2];
    __shared__ float s_e[POOL_CHUNK];

    float acc0 = 0.f, acc1 = 0.f, acc2 = 0.f;
    if (end > start) {
        if (tid == 0) {
            float m = -1e30f;
            for (int i = start; i < end; ++i) m = fmaxf(m, att[i]);
            float den = 0.f;
            for (int i = start; i < end; ++i) den += __expf(att[i] - m);
            s_md[0] = m;
            s_md[1] = den;
        }
        __syncthreads();
        const float m       = s_md[0];
        const float inv_den = 1.0f / s_md[1];       // den >= exp(0) = 1 when non-empty

        for (int cs = start; cs < end; cs += POOL_CHUNK) {
            const int cn = min(POOL_CHUNK, end - cs);
            __syncthreads();
            if (tid < cn) s_e[tid] = __expf(att[cs + tid] - m);
            __syncthreads();
            for (int j = 0; j < cn; ++j) {
                const float e = s_e[j];
                const float* row = x + (size_t)(cs + j) * DIM;
                acc0 += e * row[tid];
                acc1 += e * row[tid + 256];
                acc2 += e * row[tid + 512];
            }
        }
        acc0 *= inv_den; acc1 *= inv_den; acc2 *= inv_den;
    }
    float* out = repre + (size_t)b * DIM;
    out[tid]       = acc0;
    out[tid + 256] = acc1;
    out[tid + 512] = acc2;
}

// ---------------------------------------------------------------------------
// Kernel 4: logits = repre @ Wpad^T + bias via V_WMMA_F32_16X16X4_F32.
// Block = 128 threads = 4 waves. Block stages a 16x768 repre tile into LDS
// (row pitch 772 floats -> bank-conflict-free ds_load_b64); each wave owns
// one 16-column tile of the 64-wide padded output. Inner loop is branch-free:
// unconditional ds_load_b64 (A) + global_load_b64 (B, L2-resident) + wmma.
//   A layout (16x4 f32): lanes 0-15 = M, K = 2*half + vgpr  -> float2 load.
//   B layout (4x16 f32): lanes 0-15 = N, K = 2*half + vgpr  -> float2 load.
//   C/D (16x16 f32):     N = lane&15, M = vgpr + 8*half.
// ---------------------------------------------------------------------------
#define A_PITCH 772   // 772 % 64 == 4 -> conflict-free across 16 rows
__global__ void gemm_kernel(const float* __restrict__ repre, const float* __restrict__ wpad,
                            const float* __restrict__ bpad, float* __restrict__ out) {
    __shared__ float sA[16 * A_PITCH];

    const int tid     = threadIdx.x;                // 128 threads
    const int m_block = blockIdx.x * 16;

    // Cooperative load: 16 rows x 768 cols = 3072 float4, 24 per thread.
    for (int v = tid; v < (16 * DIM) / 4; v += 128) {
        const int row = (v * 4) / DIM;
        const int col = (v * 4) % DIM;
        float4 d = *(const float4*)(repre + (size_t)(m_block + row) * DIM + col);
        float* dst = &sA[row * A_PITCH + col];
        dst[0] = d.x; dst[1] = d.y; dst[2] = d.z; dst[3] = d.w;
    }
    __syncthreads();

    const int wave = tid >> 5;                      // 0..3 -> N tile
    const int lane = tid & 31;
    const int half = lane >> 4;
    const int l    = lane & 15;
    const int n    = wave * 16 + l;                 // 0..63, always valid in wpad
    const float* wrow = wpad + (size_t)n * DIM;

    v8f acc = {};
#pragma unroll 4
    for (int k0 = 0; k0 < DIM; k0 += 4) {
        const int k = k0 + 2 * half;
        v2f a;
        a.x = sA[l * A_PITCH + k];
        a.y = sA[l * A_PITCH + k + 1];
        v2f bb;
        bb.x = wrow[k];
        bb.y = wrow[k + 1];
        acc = __builtin_amdgcn_wmma_f32_16x16x4_f32(
            /*neg_a=*/false, a, /*neg_b=*/false, bb,
            /*c_mod=*/(short)0, acc, /*reuse_a=*/false, /*reuse_b=*/false);
    }

    if (n < NCLS) {
        const float bv = bpad[n];
#pragma unroll
        for (int r = 0; r < 8; ++r) {
            const int m = m_block + r + 8 * half;
            out[(size_t)m * NCLS + n] = acc[r] + bv;
        }
    }
}

// ---------------------------------------------------------------------------
// Launch
// ---------------------------------------------------------------------------
extern "C" void kernel_launch(void* const* d_in, const int* in_sizes, int n_in,
                              void* d_out, int out_size, void* d_ws, size_t ws_size,
                              hipStream_t stream) {
    const float* x    = (const float*)d_in[0];
    const int*   lab  = (const int*)  d_in[1];
    const int*   seg  = (const int*)  d_in[2];
    const float* W    = (const float*)d_in[3];
    const float* bias = (const float*)d_in[4];
    float*       out  = (float*)d_out;

    // Workspace layout (all 4-byte typed; offsets keep 16B alignment):
    float* att       = (float*)d_ws;                 // N floats
    int*   seg_start = (int*)(att + N_SENT);         // B ints
    int*   seg_end   = seg_start + NBAG;             // B ints
    float* repre     = (float*)(seg_end + NBAG);     // B*D floats
    float* wpad      = repre + (size_t)NBAG * DIM;   // 64*D floats
    float* bpad      = wpad + (size_t)NPAD * DIM;    // 64 floats

    att_kernel       <<<N_SENT / 8,    256, 0, stream>>>(x, lab, W, att);
    seg_init_kernel  <<<NBAG / 256,    256, 0, stream>>>(seg_start, seg_end);
    seg_bounds_kernel<<<N_SENT / 256,  256, 0, stream>>>(seg, seg_start, seg_end);
    wpad_kernel      <<<NPAD,          256, 0, stream>>>(W, bias, wpad, bpad);
    pool_kernel      <<<NBAG,          256, 0, stream>>>(x, att, seg_start, seg_end, repre);
    gemm_kernel      <<<NBAG / 16,     128, 0, stream>>>(repre, wpad, bpad, out);
}